// GATConv_893353198520
// MI455X (gfx1250) — compile-verified
//
#include <hip/hip_runtime.h>
#include <hip/hip_bf16.h>
#include <math.h>

typedef __attribute__((ext_vector_type(2))) float v2f;
typedef __attribute__((ext_vector_type(8))) float v8f;

#define IN_FEATS 256
#define HEADS 4
#define OUT_FEATS 32
#define HF (HEADS * OUT_FEATS)   // 128
#define NEG_SLOPE 0.2f

// ---------------------------------------------------------------------------
// Atomic helpers: native f32 add + bit-trick float max
// ---------------------------------------------------------------------------
__device__ __forceinline__ void atomicAddF(float* addr, float v) {
    __hip_atomic_fetch_add(addr, v, __ATOMIC_RELAXED, __HIP_MEMORY_SCOPE_AGENT);
}

__device__ __forceinline__ void atomicMaxF(float* addr, float val) {
    // Works for mixed-sign floats given init = -inf (0xFF800000).
    if (val >= 0.0f) {
        atomicMax((int*)addr, __float_as_int(val));
    } else {
        atomicMin((unsigned int*)addr, __float_as_uint(val));
    }
}

// ---------------------------------------------------------------------------
// Kernel 1: ft[N,128] = feat[N,256] @ W[128,256]^T   via V_WMMA_F32_16X16X4_F32
// One wave computes a 16(M) x 16(N) tile; 8 waves/block cover all 128 columns.
// A tile 16x4 f32: lanes 0-15 = M rows, VGPR j = K (lane>>4)*2 + j
// B tile 4x16 f32: lanes 0-15 = N cols, VGPR j = K (lane>>4)*2 + j  (B = W^T)
// ---------------------------------------------------------------------------
__global__ __launch_bounds__(256) void gat_gemm_wmma(
    const float* __restrict__ feat, const float* __restrict__ W,
    float* __restrict__ ft, int N) {
    const int wave  = threadIdx.x >> 5;
    const int lane  = threadIdx.x & 31;
    const int lrow  = lane & 15;
    const int khalf = lane >> 4;
    const int m0 = blockIdx.x * 16;
    const int n0 = wave * 16;
    if (m0 >= N) return;

    const float* aptr = feat + (size_t)(m0 + lrow) * IN_FEATS + 2 * khalf;
    const float* bptr = W    + (size_t)(n0 + lrow) * IN_FEATS + 2 * khalf;

    v8f acc = {};
#pragma unroll 8
    for (int k = 0; k < IN_FEATS; k += 4) {
        v2f a = *(const v2f*)(aptr + k);
        v2f b = *(const v2f*)(bptr + k);
        acc = __builtin_amdgcn_wmma_f32_16x16x4_f32(
            /*neg_a=*/false, a, /*neg_b=*/false, b,
            /*c_mod=*/(short)0, acc, /*reuse_a=*/false, /*reuse_b=*/false);
    }

    // D layout: VGPR r -> row m0 + r + 8*khalf, col n0 + lrow
    float* out = ft + (size_t)(m0 + 8 * khalf) * HF + n0 + lrow;
#pragma unroll
    for (int r = 0; r < 8; ++r) {
        out[(size_t)r * HF] = acc[r];
    }
}

// ---------------------------------------------------------------------------
// Kernel 2: per-(node,head) attention projections el/er
// ---------------------------------------------------------------------------
__global__ void gat_scores(const float* __restrict__ ft,
                           const float* __restrict__ attn_l,
                           const float* __restrict__ attn_r,
                           float* __restrict__ el, float* __restrict__ er,
                           int NH) {
    int t = blockIdx.x * blockDim.x + threadIdx.x;
    if (t >= NH) return;
    int h = t & (HEADS - 1);
    int n = t >> 2;
    const float4* f  = (const float4*)(ft + (size_t)n * HF + h * OUT_FEATS);
    const float4* al = (const float4*)(attn_l + h * OUT_FEATS);
    const float4* ar = (const float4*)(attn_r + h * OUT_FEATS);
    float sl = 0.0f, sr = 0.0f;
#pragma unroll
    for (int i = 0; i < OUT_FEATS / 4; ++i) {
        float4 v = f[i];
        float4 a = al[i];
        float4 b = ar[i];
        sl += v.x * a.x + v.y * a.y + v.z * a.z + v.w * a.w;
        sr += v.x * b.x + v.y * b.y + v.z * b.z + v.w * b.w;
    }
    el[t] = sl;
    er[t] = sr;
}

// ---------------------------------------------------------------------------
// Kernel 3a: init output to bias ; 3b: init emax/esum
// ---------------------------------------------------------------------------
__global__ void gat_init_out(float* __restrict__ out,
                             const float* __restrict__ bias, int total) {
    int t = blockIdx.x * blockDim.x + threadIdx.x;
    if (t < total) out[t] = bias[t & (HF - 1)];
}

__global__ void gat_init_stats(float* __restrict__ emax,
                               float* __restrict__ esum, int nh) {
    int t = blockIdx.x * blockDim.x + threadIdx.x;
    if (t < nh) {
        emax[t] = __int_as_float(0xFF800000);  // -inf
        esum[t] = 0.0f;
    }
}

// ---------------------------------------------------------------------------
// Kernel 4: edge pass 1 — leaky score, atomic max per (dst, head)
// ---------------------------------------------------------------------------
__global__ void gat_edge_max(const int* __restrict__ src,
                             const int* __restrict__ dst,
                             const float* __restrict__ el,
                             const float* __restrict__ er,
                             float* __restrict__ emax, int EH) {
    int t = blockIdx.x * blockDim.x + threadIdx.x;
    if (t >= EH) return;
    int h = t & (HEADS - 1);
    int e = t >> 2;
    int s = src[e], d = dst[e];
    float sc = el[s * HEADS + h] + er[d * HEADS + h];
    sc = (sc >= 0.0f) ? sc : NEG_SLOPE * sc;
    atomicMaxF(&emax[d * HEADS + h], sc);
}

// ---------------------------------------------------------------------------
// Kernel 5: edge pass 2 — exp(e - emax[dst]) atomic-summed per (dst, head)
// ---------------------------------------------------------------------------
__global__ void gat_edge_sum(const int* __restrict__ src,
                             const int* __restrict__ dst,
                             const float* __restrict__ el,
                             const float* __restrict__ er,
                             const float* __restrict__ emax,
                             float* __restrict__ esum, int EH) {
    int t = blockIdx.x * blockDim.x + threadIdx.x;
    if (t >= EH) return;
    int h = t & (HEADS - 1);
    int e = t >> 2;
    int s = src[e], d = dst[e];
    float sc = el[s * HEADS + h] + er[d * HEADS + h];
    sc = (sc >= 0.0f) ? sc : NEG_SLOPE * sc;
    float ex = __expf(sc - emax[d * HEADS + h]);
    atomicAddF(&esum[d * HEADS + h], ex);
}

// ---------------------------------------------------------------------------
// Kernel 6: aggregation — one wave per edge; lane l handles float4 at l*4
// (all 4 elements share head h = l>>3). Recompute a deterministically.
// ---------------------------------------------------------------------------
__global__ __launch_bounds__(256) void gat_aggregate(
    const int* __restrict__ src, const int* __restrict__ dst,
    const float* __restrict__ ft, const float* __restrict__ el,
    const float* __restrict__ er, const float* __restrict__ emax,
    const float* __restrict__ esum, float* __restrict__ out, int E) {
    int e = blockIdx.x * (blockDim.x >> 5) + (threadIdx.x >> 5);
    if (e >= E) return;
    int lane = threadIdx.x & 31;
    int s = src[e], d = dst[e];
    int h = lane >> 3;  // lane*4 .. lane*4+3 all inside head (lane*4)/32

    float sc = el[s * HEADS + h] + er[d * HEADS + h];
    sc = (sc >= 0.0f) ? sc : NEG_SLOPE * sc;
    float ex = __expf(sc - emax[d * HEADS + h]);
    float a  = ex / fmaxf(esum[d * HEADS + h], 1e-16f);

    float4 v = *(const float4*)(ft + (size_t)s * HF + lane * 4);
    float* o = out + (size_t)d * HF + lane * 4;
    atomicAddF(o + 0, v.x * a);
    atomicAddF(o + 1, v.y * a);
    atomicAddF(o + 2, v.z * a);
    atomicAddF(o + 3, v.w * a);
}

// ---------------------------------------------------------------------------
extern "C" void kernel_launch(void* const* d_in, const int* in_sizes, int n_in,
                              void* d_out, int out_size, void* d_ws, size_t ws_size,
                              hipStream_t stream) {
    const float* feat   = (const float*)d_in[0];
    const int*   src    = (const int*)d_in[1];
    const int*   dst    = (const int*)d_in[2];
    const float* W      = (const float*)d_in[3];
    const float* attn_l = (const float*)d_in[4];
    const float* attn_r = (const float*)d_in[5];
    const float* bias   = (const float*)d_in[6];
    float* out = (float*)d_out;

    const int N = in_sizes[0] / IN_FEATS;  // 100000
    const int E = in_sizes[1];             // 1600000

    // Workspace layout (all fp32, 256B-aligned sections)
    char* ws = (char*)d_ws;
    float* ft   = (float*)ws;                                   // N*128
    float* el   = (float*)(ws + ((size_t)N * HF * 4 + 255 & ~(size_t)255));
    float* er   = el + (size_t)N * HEADS;
    float* emax = er + (size_t)N * HEADS;
    float* esum = emax + (size_t)N * HEADS;

    const int NH = N * HEADS;
    const int EH = E * HEADS;

    // 1) WMMA projection
    int mblocks = (N + 15) / 16;
    gat_gemm_wmma<<<mblocks, 256, 0, stream>>>(feat, W, ft, N);

    // 2) el / er
    gat_scores<<<(NH + 255) / 256, 256, 0, stream>>>(ft, attn_l, attn_r, el, er, NH);

    // 3) init
    gat_init_out<<<((size_t)N * HF + 255) / 256, 256, 0, stream>>>(out, bias, N * HF);
    gat_init_stats<<<(NH + 255) / 256, 256, 0, stream>>>(emax, esum, NH);

    // 4) segment max
    gat_edge_max<<<(EH + 255) / 256, 256, 0, stream>>>(src, dst, el, er, emax, EH);

    // 5) segment sum of exp
    gat_edge_sum<<<(EH + 255) / 256, 256, 0, stream>>>(src, dst, el, er, emax, esum, EH);

    // 6) weighted scatter aggregation (one wave per edge)
    gat_aggregate<<<(E + 7) / 8, 256, 0, stream>>>(src, dst, ft, el, er, emax, esum, out, E);
}